// ReluTransformer_89446988907158
// MI455X (gfx1250) — compile-verified
//
#include <hip/hip_runtime.h>

typedef __attribute__((ext_vector_type(2))) float v2f;
typedef __attribute__((ext_vector_type(8))) float v8f;

// Select element k (0..7) of an 8-wide f32 accumulator without scratch:
// lowers to a short v_cndmask chain.
__device__ __forceinline__ float pick8(v8f c, int k) {
  float r = c[0];
  r = (k == 1) ? c[1] : r;
  r = (k == 2) ? c[2] : r;
  r = (k == 3) ? c[3] : r;
  r = (k == 4) ? c[4] : r;
  r = (k == 5) ? c[5] : r;
  r = (k == 6) ? c[6] : r;
  r = (k == 7) ? c[7] : r;
  return r;
}

// One wave32 handles one 16-element tile. For each tile we issue two
// V_WMMA_F32_16X16X4_F32 ops:
//   D = A(16x4) x B(4x16), A row i = [max(w_i,0), min(w_i,0), bias_i, 0],
//   B col i = [x_i, y_i, 1, 0]  =>  D[i][i] = the CROWN-style relaxed bound.
// f32 A/B layout (ISA 7.12.2): lanes 0-15 carry K=0 (v.x) and K=1 (v.y);
// lanes 16-31 carry K=2 (v.x) and K=3 (v.y). C/D: VGPR r, lanes 0-15 ->
// (M=r, N=lane); lanes 16-31 -> (M=r+8, N=lane-16).
__global__ void relu_bounds_wmma_kernel(const float* __restrict__ lb_in,
                                        const float* __restrict__ ub_in,
                                        const float* __restrict__ al_in,
                                        float* __restrict__ out, int n) {
  const int tid   = blockIdx.x * blockDim.x + threadIdx.x;
  const int lane  = threadIdx.x & 31;
  const int wave  = tid >> 5;
  const int tiles = (n + 15) >> 4;
  if (wave >= tiles) return;  // wave-uniform: EXEC stays all-ones for WMMA

  const int e0  = wave << 4;
  int idx  = e0 + (lane & 15);
  int idxc = (idx < n) ? idx : (n - 1);  // clamp loads; stores predicated later

  const float lb = lb_in[idxc];
  const float ub = ub_in[idxc];
  const float al = al_in[idxc];

  // ReLU relaxation coefficients (negative_slope == 0 branch).
  const float a   = fminf(fmaxf(al, 0.0f), 1.0f);   // clamp_ste forward value
  const float su  = ub / (ub - lb);                 // slope_upper (ub > lb guaranteed)
  const bool pos      = (lb >= 0.0f) && (ub >= 0.0f);
  const bool cross    = (lb <  0.0f) && (ub >= 0.0f);
  const bool cross_s  = (lb <  0.0f) && (ub >  0.0f);
  const float wl = (pos ? 1.0f : 0.0f) + (cross ? a  : 0.0f);
  const float wu = (pos ? 1.0f : 0.0f) + (cross ? su : 0.0f);
  const float bias_u = cross_s ? (-su * lb) : 0.0f;

  const bool lo = (lane < 16);

  // lower = max(wl,0)*lb + min(wl,0)*ub + 0
  v2f A_lo, B_lo, A_up, B_up;
  A_lo.x = lo ? fmaxf(wl, 0.0f) : 0.0f;   // K0 : pos part   | K2 : bias_lower = 0
  A_lo.y = lo ? fminf(wl, 0.0f) : 0.0f;   // K1 : neg part   | K3 : 0
  B_lo.x = lo ? lb : 1.0f;                // K0 : lb         | K2 : 1 (bias lane)
  B_lo.y = lo ? ub : 0.0f;                // K1 : ub         | K3 : 0

  // upper = max(wu,0)*ub + min(wu,0)*lb + bias_upper
  A_up.x = lo ? fmaxf(wu, 0.0f) : bias_u;
  A_up.y = lo ? fminf(wu, 0.0f) : 0.0f;
  B_up.x = lo ? ub : 1.0f;
  B_up.y = lo ? lb : 0.0f;

  v8f zero = {};
  // 8 args: (neg_a, A, neg_b, B, c_mod, C, reuse_a, reuse_b)
  v8f c_lo = __builtin_amdgcn_wmma_f32_16x16x4_f32(
      false, A_lo, false, B_lo, (short)0, zero, false, false);
  v8f c_up = __builtin_amdgcn_wmma_f32_16x16x4_f32(
      false, A_up, false, B_up, (short)0, zero, false, false);

  // Diagonal extraction: element i<8 at (VGPR i, lane i); element i>=8 at
  // (VGPR i-8, lane i+16).
  int di = -1, vr = 0;
  if (lane < 8)        { di = e0 + lane;      vr = lane;      }
  else if (lane >= 24) { di = e0 + lane - 16; vr = lane - 24; }

  if (di >= 0 && di < n) {
    out[di]     = pick8(c_lo, vr);  // lower, first n outputs
    out[n + di] = pick8(c_up, vr);  // upper, second n outputs
  }
}

extern "C" void kernel_launch(void* const* d_in, const int* in_sizes, int n_in,
                              void* d_out, int out_size, void* d_ws, size_t ws_size,
                              hipStream_t stream) {
  (void)n_in; (void)out_size; (void)d_ws; (void)ws_size;
  const float* lb = (const float*)d_in[0];  // lower_bound_prev
  const float* ub = (const float*)d_in[1];  // upper_bound_prev
  const float* al = (const float*)d_in[2];  // alpha
  float* out = (float*)d_out;               // [lower (n) | upper (n)]

  const int n = in_sizes[0];                // 8192
  const int tiles = (n + 15) / 16;          // one wave32 per 16-elem tile
  const int threads = 256;                  // 8 waves per workgroup
  const int total = tiles * 32;
  const int blocks = (total + threads - 1) / threads;
  relu_bounds_wmma_kernel<<<blocks, threads, 0, stream>>>(lb, ub, al, out, n);
}